// BayesianLayer_163208757903
// MI455X (gfx1250) — compile-verified
//
#include <hip/hip_runtime.h>
#include <hip/hip_bf16.h>
#include <cstdint>
#include <cstddef>

#define B_   64
#define IN_  1024
#define OUT_ 1024
#define ICH  4     // i-chunks in the eps streaming kernel

typedef __attribute__((ext_vector_type(2))) float v2f;
typedef __attribute__((ext_vector_type(4))) float v4f;
typedef __attribute__((ext_vector_type(8))) float v8f;

__device__ __forceinline__ float softplus_f(float v) {
    // numerically stable softplus: max(v,0) + log1p(exp(-|v|))
    return fmaxf(v, 0.0f) + log1pf(expf(-fabsf(v)));
}

// ---------------------------------------------------------------------------
// Kernel 1: sigma = softplus(ro) over IN_*OUT_, float4 vectorized.
// Precomputed once so the streaming kernel doesn't redo 64M exp/log pairs.
// ---------------------------------------------------------------------------
__global__ __launch_bounds__(256) void k_softplus(const float* __restrict__ ro,
                                                  float* __restrict__ sigma) {
    int idx = (blockIdx.x * 256 + threadIdx.x) * 4;
    v4f r = *(const v4f*)(ro + idx);
    v4f s;
    s.x = softplus_f(r.x);
    s.y = softplus_f(r.y);
    s.z = softplus_f(r.z);
    s.w = softplus_f(r.w);
    *(v4f*)(sigma + idx) = s;
}

// ---------------------------------------------------------------------------
// Kernel 2: out[b,o] = (x @ mu)[b,o] + eps_bias[b,o]*softplus(ro_bias[o]) + mu_bias[o]
// One wave per 16x16 output tile, V_WMMA_F32_16X16X4_F32, K stepped by 4.
//
// ISA VGPR layouts (CDNA5 ISA §7.12.2):
//   A (16x4 f32): lanes 0-15 -> M=lane, VGPR{0,1}=K{0,1}; lanes 16-31 -> K{2,3}
//   B (4x16 f32): mirrored — lanes 0-15 -> N=lane, VGPR{0,1}=K{0,1}; lanes 16-31 -> K{2,3}
//   C/D (16x16 f32): VGPR r, lanes 0-15 -> M=r, N=lane; lanes 16-31 -> M=r+8
// ---------------------------------------------------------------------------
__global__ __launch_bounds__(32) void k_wmma_mu_bias(const float* __restrict__ x,
                                                     const float* __restrict__ mu,
                                                     const float* __restrict__ mu_bias,
                                                     const float* __restrict__ ro_bias,
                                                     const float* __restrict__ eps_bias,
                                                     float* __restrict__ out) {
    const int tile    = blockIdx.x;          // 0 .. (B_/16)*(OUT_/16)-1 = 255
    const int tM      = tile >> 6;           // tile / (OUT_/16)
    const int tN      = tile & 63;           // tile % (OUT_/16)
    const int lane    = threadIdx.x & 31;
    const int half    = lane >> 4;           // 0: K pair {0,1}; 1: K pair {2,3}
    const int l       = lane & 15;
    const int rowBase = tM * 16;
    const int colBase = tN * 16;

    v8f acc = {};
    const float* arow = x + (size_t)(rowBase + l) * IN_;

    for (int k0 = 0; k0 < IN_; k0 += 4) {
        const int ka = k0 + 2 * half;
        // A: x rows are contiguous in K -> one b64 load per lane
        v2f a = *(const v2f*)(arow + ka);
        // B: mu rows are contiguous in N, so K is strided -> two b32 loads
        v2f b;
        b.x = mu[(size_t)ka * OUT_ + colBase + l];
        b.y = mu[(size_t)(ka + 1) * OUT_ + colBase + l];
        acc = __builtin_amdgcn_wmma_f32_16x16x4_f32(
            /*neg_a=*/false, a, /*neg_b=*/false, b,
            /*c_mod=*/(short)0, acc, /*reuse_a=*/false, /*reuse_b=*/false);
    }

    const int o  = colBase + l;
    const float sb = softplus_f(ro_bias[o]);
    const float mb = mu_bias[o];
#pragma unroll
    for (int r = 0; r < 8; ++r) {
        const int row = rowBase + r + 8 * half;
        out[(size_t)row * OUT_ + o] = acc[r] + eps_bias[(size_t)row * OUT_ + o] * sb + mb;
    }
}

// ---------------------------------------------------------------------------
// Kernel 3 (the bandwidth kernel): partial[ic][b][o] = sum_{i in chunk} x[b,i]*sigma[i,o]*eps[b,i,o]
// Grid (B_, ICH), 256 threads/block (8 waves); each thread owns 4 consecutive o
// -> b128 loads of eps and sigma, fully coalesced (the block covers an entire
// contiguous 4 KB row eps[b,i,:] per i step). x[b,i] is wave-uniform -> s_load.
// sigma/mu-sized operands are L2-resident (8 MB << 192 MB); eps streams from HBM.
// ---------------------------------------------------------------------------
__global__ __launch_bounds__(256) void k_stream_eps(const float* __restrict__ x,
                                                    const float* __restrict__ eps,
                                                    const float* __restrict__ sigma,
                                                    float* __restrict__ partial) {
    const int b    = blockIdx.x;            // 0..B_-1
    const int ic   = blockIdx.y;            // 0..ICH-1
    const int o    = threadIdx.x * 4;       // 0..1020
    const int iBeg = ic * (IN_ / ICH);
    const int iEnd = iBeg + (IN_ / ICH);

    v4f acc = {0.0f, 0.0f, 0.0f, 0.0f};
    const float* xrow = x + (size_t)b * IN_;

#pragma unroll 4
    for (int i = iBeg; i < iEnd; ++i) {
        const float xv = xrow[i];                                   // uniform -> SGPR
        v4f e = *(const v4f*)(eps + ((size_t)(b * IN_ + i)) * OUT_ + o);
        v4f s = *(const v4f*)(sigma + (size_t)i * OUT_ + o);
        acc += (xv * s) * e;
    }

    *(v4f*)(partial + ((size_t)(ic * B_ + b)) * OUT_ + o) = acc;
}

// ---------------------------------------------------------------------------
// Kernel 4: out += sum over ICH partials. 64K elements, float4.
// ---------------------------------------------------------------------------
__global__ __launch_bounds__(256) void k_reduce(const float* __restrict__ partial,
                                                float* __restrict__ out) {
    int idx = (blockIdx.x * 256 + threadIdx.x) * 4;   // over B_*OUT_
    v4f a = *(v4f*)(out + idx);
#pragma unroll
    for (int c = 0; c < ICH; ++c)
        a += *(const v4f*)(partial + (size_t)c * B_ * OUT_ + idx);
    *(v4f*)(out + idx) = a;
}

// ---------------------------------------------------------------------------
extern "C" void kernel_launch(void* const* d_in, const int* in_sizes, int n_in,
                              void* d_out, int out_size, void* d_ws, size_t ws_size,
                              hipStream_t stream) {
    const float* x        = (const float*)d_in[0]; // [B, IN]
    const float* mu       = (const float*)d_in[1]; // [IN, OUT]
    const float* ro       = (const float*)d_in[2]; // [IN, OUT]
    const float* mu_bias  = (const float*)d_in[3]; // [1, OUT]
    const float* ro_bias  = (const float*)d_in[4]; // [1, OUT]
    const float* eps      = (const float*)d_in[5]; // [B, IN, OUT]
    const float* eps_bias = (const float*)d_in[6]; // [B, OUT]
    float* out = (float*)d_out;                    // [B, OUT]

    // workspace layout: sigma (IN_*OUT_ floats = 4 MB) | partials (ICH*B_*OUT_ floats = 1 MB)
    float* sigma   = (float*)d_ws;
    float* partial = sigma + (size_t)IN_ * OUT_;

    // 1) sigma = softplus(ro)
    k_softplus<<<(IN_ * OUT_) / (256 * 4), 256, 0, stream>>>(ro, sigma);

    // 2) out = x @ mu + bias   (WMMA f32)
    k_wmma_mu_bias<<<(B_ / 16) * (OUT_ / 16), 32, 0, stream>>>(
        x, mu, mu_bias, ro_bias, eps_bias, out);

    // 3) partials of sum_i x * sigma * eps   (HBM-bound streaming, 256 MB)
    k_stream_eps<<<dim3(B_, ICH), 256, 0, stream>>>(x, eps, sigma, partial);

    // 4) out += partials
    k_reduce<<<(B_ * OUT_) / (256 * 4), 256, 0, stream>>>(partial, out);
}